// CRDLoss_63136019251442
// MI455X (gfx1250) — compile-verified
//
#include <hip/hip_runtime.h>
#include <math.h>

typedef __attribute__((ext_vector_type(2))) float v2f;
typedef __attribute__((ext_vector_type(8))) float v8f;

// ---------------------------------------------------------------------------
// Kernel 1: gather memory rows + WMMA f32 dot-product scoring.
// Grid: (ceil(tilesPerB/8), B). Block: 256 threads = 8 waves (wave32).
// Each wave computes 16 row-scores for both memory banks via
// V_WMMA_F32_16X16X4_F32 (full fp32, no precision loss).
//   S1[b,k] = dot(memory_v2[full_idx[b,k]], f_s[b])   (-> out_v1)
//   S2[b,k] = dot(memory_v1[full_idx[b,k]], f_t[b])   (-> out_v2)
// ---------------------------------------------------------------------------
__global__ __launch_bounds__(256)
void k_gather_score(const float* __restrict__ f_s, const float* __restrict__ f_t,
                    const float* __restrict__ mem1, const float* __restrict__ mem2,
                    const int* __restrict__ idx, const int* __restrict__ cidx,
                    float* __restrict__ S1, float* __restrict__ S2,
                    int K, int tilesPerB)
{
    const int D = 128;
    const int b = blockIdx.y;
    const int K1 = K + 1;

    __shared__ __align__(16) float fS[128];
    __shared__ __align__(16) float fT[128];
    const int tid = threadIdx.x;
    if (tid < 128)      fS[tid]       = f_s[b * D + tid];
    else                fT[tid - 128] = f_t[b * D + (tid - 128)];
    __syncthreads();

    const int wave = tid >> 5;
    const int lane = tid & 31;
    const int tile = blockIdx.x * 8 + wave;
    if (tile >= tilesPerB) return;           // wave-uniform exit: EXEC stays all-1 for live waves

    const int kbase = tile * 16;
    const int m    = lane & 15;              // row within 16x16 tile (A-matrix M index)
    const int half = lane >> 4;              // selects K pair {0,1} vs {2,3}
    int k  = kbase + m;
    int kc = (k <= K) ? k : K;               // clamp tail rows (results discarded)
    const int row = (kc == 0) ? idx[b] : cidx[(size_t)b * K + (kc - 1)];

    const v2f* __restrict__ w1 = (const v2f*)(mem1 + (size_t)row * D);
    const v2f* __restrict__ w2 = (const v2f*)(mem2 + (size_t)row * D);
    const v2f* __restrict__ pS = (const v2f*)fS;
    const v2f* __restrict__ pT = (const v2f*)fT;

    v8f accS = {};   // dots vs memory_v2 / f_s
    v8f accT = {};   // dots vs memory_v1 / f_t

    // D = 128 -> 32 steps of K=4. float2 index = kk*2 + half
    // (A: lane%16 = M, vgpr pair = K slice; B: feature vec replicated over N).
    #pragma unroll 8
    for (int kk = 0; kk < 32; ++kk) {
        const int c2 = kk * 2 + half;
        v2f a1 = w1[c2];
        v2f a2 = w2[c2];
        v2f bT = pT[c2];
        v2f bS = pS[c2];
        accT = __builtin_amdgcn_wmma_f32_16x16x4_f32(false, a1, false, bT,
                                                     (short)0, accT, false, false);
        accS = __builtin_amdgcn_wmma_f32_16x16x4_f32(false, a2, false, bS,
                                                     (short)0, accS, false, false);
    }

    // C/D layout: lane 0 holds rows 0..7 (elems 0..7), lane 16 holds rows 8..15.
    // All 16 output columns are identical (B columns replicated), so lanes 0/16 suffice.
    if (m == 0) {
        const int rbase = kbase + half * 8;
        #pragma unroll
        for (int e = 0; e < 8; ++e) {
            const int kk2 = rbase + e;
            if (kk2 <= K) {
                S1[(size_t)b * K1 + kk2] = accS[e];
                S2[(size_t)b * K1 + kk2] = accT[e];
            }
        }
    }
}

// ---------------------------------------------------------------------------
// Kernel 2: per-block partial sums of exp(dot/T) for both score arrays.
// ---------------------------------------------------------------------------
__global__ __launch_bounds__(256)
void k_sumexp(const float* __restrict__ S1, const float* __restrict__ S2,
              float* __restrict__ P1, float* __restrict__ P2, int n, float invT)
{
    __shared__ float sh1[256];
    __shared__ float sh2[256];
    const int tid = threadIdx.x;
    float s1 = 0.f, s2 = 0.f;
    for (int i = blockIdx.x * blockDim.x + tid; i < n; i += gridDim.x * blockDim.x) {
        s1 += expf(S1[i] * invT);
        s2 += expf(S2[i] * invT);
    }
    sh1[tid] = s1; sh2[tid] = s2;
    __syncthreads();
    for (int o = 128; o > 0; o >>= 1) {
        if (tid < o) { sh1[tid] += sh1[tid + o]; sh2[tid] += sh2[tid + o]; }
        __syncthreads();
    }
    if (tid == 0) { P1[blockIdx.x] = sh1[0]; P2[blockIdx.x] = sh2[0]; }
}

// ---------------------------------------------------------------------------
// Kernel 3: finalize Z_v1 / Z_v2 = mean(exp) * n_data.  One block.
// ---------------------------------------------------------------------------
__global__ __launch_bounds__(256)
void k_zfinal(const float* __restrict__ P1, const float* __restrict__ P2,
              float* __restrict__ Z, float zscale)
{
    __shared__ float sh[512];
    const int tid = threadIdx.x;
    sh[tid]       = P1[tid];
    sh[256 + tid] = P2[tid];
    __syncthreads();
    for (int o = 128; o > 0; o >>= 1) {
        if (tid < o) { sh[tid] += sh[tid + o]; sh[256 + tid] += sh[256 + tid + o]; }
        __syncthreads();
    }
    if (tid == 0) { Z[0] = sh[0] * zscale; Z[1] = sh[256] * zscale; }
}

// ---------------------------------------------------------------------------
// Kernel 4: per-block partial sums of the NCE loss terms (both branches).
//   k==0: log(x/(x+c)); k>0: log(mPn/(x+c)) with x = exp(dot/T)/Z.
// ---------------------------------------------------------------------------
__global__ __launch_bounds__(256)
void k_losspartial(const float* __restrict__ S1, const float* __restrict__ S2,
                   const float* __restrict__ Z, float* __restrict__ Q,
                   int n, int K1, float invT, float mPn, float cadd)
{
    __shared__ float sh[256];
    const int tid = threadIdx.x;
    const float Z1 = Z[0];
    const float Z2 = Z[1];
    float acc = 0.f;
    for (int i = blockIdx.x * blockDim.x + tid; i < n; i += gridDim.x * blockDim.x) {
        const int k = i % K1;
        const float x1 = expf(S1[i] * invT) / Z1;
        const float x2 = expf(S2[i] * invT) / Z2;
        if (k == 0)
            acc += logf(x1 / (x1 + cadd)) + logf(x2 / (x2 + cadd));
        else
            acc += logf(mPn / (x1 + cadd)) + logf(mPn / (x2 + cadd));
    }
    sh[tid] = acc;
    __syncthreads();
    for (int o = 128; o > 0; o >>= 1) {
        if (tid < o) sh[tid] += sh[tid + o];
        __syncthreads();
    }
    if (tid == 0) Q[blockIdx.x] = sh[0];
}

// ---------------------------------------------------------------------------
// Kernel 5: final loss = -(sum of terms)/B  -> d_out[0].  One block.
// ---------------------------------------------------------------------------
__global__ __launch_bounds__(256)
void k_lossfinal(const float* __restrict__ Q, float* __restrict__ out, int B)
{
    __shared__ float sh[256];
    const int tid = threadIdx.x;
    sh[tid] = Q[tid];
    __syncthreads();
    for (int o = 128; o > 0; o >>= 1) {
        if (tid < o) sh[tid] += sh[tid + o];
        __syncthreads();
    }
    if (tid == 0) out[0] = -sh[0] / (float)B;
}

// ---------------------------------------------------------------------------
// Kernel 6: momentum update of the 256 touched rows (banks already copied
// into d_out by async D2D memcpy). Duplicate idx: largest b wins (determinate).
// ---------------------------------------------------------------------------
__global__ __launch_bounds__(128)
void k_update(const float* __restrict__ mem1, const float* __restrict__ mem2,
              const float* __restrict__ f_s, const float* __restrict__ f_t,
              const int* __restrict__ idx,
              float* __restrict__ out1, float* __restrict__ out2, int B)
{
    const int D = 128;
    const int b = blockIdx.x;
    const int t = threadIdx.x;
    const int row = idx[b];
    for (int b2 = b + 1; b2 < B; ++b2)
        if (idx[b2] == row) return;          // a later write would win; skip ours

    const float n1 = mem1[(size_t)row * D + t] * 0.5f + f_s[b * D + t] * 0.5f;
    const float n2 = mem2[(size_t)row * D + t] * 0.5f + f_t[b * D + t] * 0.5f;

    __shared__ float sh[256];
    sh[t]       = n1 * n1;
    sh[128 + t] = n2 * n2;
    __syncthreads();
    for (int o = 64; o > 0; o >>= 1) {
        if (t < o) { sh[t] += sh[t + o]; sh[128 + t] += sh[128 + t + o]; }
        __syncthreads();
    }
    out1[(size_t)row * D + t] = n1 / sqrtf(sh[0]);
    out2[(size_t)row * D + t] = n2 / sqrtf(sh[128]);
}

// ---------------------------------------------------------------------------
extern "C" void kernel_launch(void* const* d_in, const int* in_sizes, int n_in,
                              void* d_out, int out_size, void* d_ws, size_t ws_size,
                              hipStream_t stream)
{
    const float* f_s  = (const float*)d_in[0];
    const float* f_t  = (const float*)d_in[1];
    const float* mem1 = (const float*)d_in[2];
    const float* mem2 = (const float*)d_in[3];
    const int*   idx  = (const int*)d_in[4];
    const int*   cidx = (const int*)d_in[5];
    float* out = (float*)d_out;

    const int B  = in_sizes[4];          // 256
    const int D  = 128;
    const int N  = in_sizes[2] / D;      // 500000
    const int K  = in_sizes[5] / B;      // 8192
    const int K1 = K + 1;
    const int n  = B * K1;               // 2,097,408 scores per bank

    // Workspace layout (floats): S1 | S2 | P1[256] | P2[256] | Q[256] | Z[2]
    float* S1 = (float*)d_ws;
    float* S2 = S1 + (size_t)n;
    float* P1 = S2 + (size_t)n;
    float* P2 = P1 + 256;
    float* Q  = P2 + 256;
    float* Z  = Q  + 256;

    const int tilesPerB = (K1 + 15) / 16;              // 513
    dim3 g1((tilesPerB + 7) / 8, B);
    k_gather_score<<<g1, 256, 0, stream>>>(f_s, f_t, mem1, mem2, idx, cidx,
                                           S1, S2, K, tilesPerB);

    const float invT = 1.0f / 0.07f;
    k_sumexp<<<256, 256, 0, stream>>>(S1, S2, P1, P2, n, invT);

    const float zscale = (float)((double)N / (double)n);   // mean * n_data
    k_zfinal<<<1, 256, 0, stream>>>(P1, P2, Z, zscale);

    const double Pn   = 1.0 / (double)N;
    const float  mPn  = (float)((double)K * Pn);
    const float  cadd = (float)((double)K * Pn + 1e-7);
    k_losspartial<<<256, 256, 0, stream>>>(S1, S2, Z, Q, n, K1, invT, mPn, cadd);
    k_lossfinal<<<1, 256, 0, stream>>>(Q, out, B);

    // Bulk bank copies into the output, then patch the updated rows.
    float* out1 = out + 1;
    float* out2 = out + 1 + (size_t)N * D;
    hipMemcpyAsync(out1, mem1, (size_t)N * D * sizeof(float),
                   hipMemcpyDeviceToDevice, stream);
    hipMemcpyAsync(out2, mem2, (size_t)N * D * sizeof(float),
                   hipMemcpyDeviceToDevice, stream);
    k_update<<<B, 128, 0, stream>>>(mem1, mem2, f_s, f_t, idx, out1, out2, B);
}